// FastBorealKernel_1571958030722
// MI455X (gfx1250) — compile-verified
//
#include <hip/hip_runtime.h>
#include <cstdint>

#define N_IN   4096
#define DEG    16
#define TPB    256
#define EPS    1.0f

// Order-preserving float -> uint key (ascending)
__device__ __forceinline__ unsigned f2key(float f) {
  unsigned u = __float_as_uint(f);
  return (u & 0x80000000u) ? ~u : (u | 0x80000000u);
}

__global__ __launch_bounds__(TPB) void boreal_kernel(
    const float* __restrict__ x,
    const float* __restrict__ w,
    const int*   __restrict__ idx,
    float*       __restrict__ out,
    int n_out) {
  __shared__ float    shX[N_IN];     // 16 KB: one batch row
  __shared__ unsigned hist[256];     // radix-select histogram
  __shared__ unsigned s_key;         // selected key prefix
  __shared__ unsigned s_rank;        // remaining rank within prefix

  const int tid = threadIdx.x;
  const int b   = blockIdx.x;
  const float* row = x + (size_t)b * N_IN;   // uniform (SGPR) base

  // ---------------- Phase A: async DMA row -> LDS ----------------
  // 4096 floats = 1024 x B128 chunks; 256 lanes x 4 rounds.
  // global_load_async_to_lds_b128: vdst = LDS byte address, vaddr = 32-bit
  // offset, saddr = 64-bit SGPR base (GVS mode). Tracked by ASYNCcnt.
  {
    unsigned ldsBase = (unsigned)(uintptr_t)&shX[0]; // low 32 bits of flat LDS ptr = LDS offset
    #pragma unroll
    for (int k = 0; k < N_IN / (TPB * 4); ++k) {
      unsigned byteOff = (unsigned)(k * TPB + tid) * 16u;
      unsigned ldsAddr = ldsBase + byteOff;
      asm volatile("global_load_async_to_lds_b128 %0, %1, %2"
                   :: "v"(ldsAddr), "v"(byteOff), "s"(row)
                   : "memory");
    }
  }
  if (tid == 0) { s_key = 0u; s_rank = (N_IN - 1) / 2; }  // rank 2047
  asm volatile("s_wait_asynccnt 0" ::: "memory");         // per-wave drain
  __syncthreads();                                        // all waves' DMA visible

  // ------- Phase B: exact rank-2047 select, MSB-first radix (4x8 bits) -------
  #pragma unroll
  for (int p = 3; p >= 0; --p) {
    hist[tid] = 0u;                 // TPB == 256 bins
    __syncthreads();
    unsigned pref = s_key;
    #pragma unroll
    for (int i = 0; i < N_IN / TPB; ++i) {
      unsigned key = f2key(shX[tid + i * TPB]);           // conflict-free stride
      bool match = (p == 3) || ((key >> ((p + 1) * 8)) == pref);
      if (match) atomicAdd(&hist[(key >> (p * 8)) & 0xFFu], 1u);  // ds_add_u32
    }
    __syncthreads();
    if (tid == 0) {
      unsigned r = s_rank, cum = 0u, c;
      for (c = 0u; c < 255u; ++c) {
        unsigned h = hist[c];
        if (cum + h > r) break;
        cum += h;
      }
      s_key  = (pref << 8) | c;
      s_rank = r - cum;
    }
    __syncthreads();
  }
  const unsigned mk = s_key;
  const float med = __uint_as_float((mk & 0x80000000u) ? (mk & 0x7FFFFFFFu) : ~mk);

  // ------- Phase C: gather + gate + max over DEG, coalesced store -------
  for (int o = tid; o < n_out; o += TPB) {
    size_t j0 = (size_t)o * DEG;
    const int4*   si4 = (const int4*)(idx + j0);
    const float4* w4  = (const float4*)(w + j0);
    if (o + TPB < n_out) {  // warm next chunk in L2 (global_prefetch_b8)
      __builtin_prefetch(idx + j0 + (size_t)TPB * DEG, 0, 3);
      __builtin_prefetch(w   + j0 + (size_t)TPB * DEG, 0, 3);
    }
    float m = -__builtin_inff();
    #pragma unroll
    for (int q = 0; q < DEG / 4; ++q) {
      int4   si = si4[q];      // B128 load, L2 resident
      float4 wv = w4[q];       // B128 load
      float a, v;
      a = shX[si.x]; v = (__builtin_fabsf(a - med) < EPS) ? (a + wv.x) : 0.0f; m = fmaxf(m, v);
      a = shX[si.y]; v = (__builtin_fabsf(a - med) < EPS) ? (a + wv.y) : 0.0f; m = fmaxf(m, v);
      a = shX[si.z]; v = (__builtin_fabsf(a - med) < EPS) ? (a + wv.z) : 0.0f; m = fmaxf(m, v);
      a = shX[si.w]; v = (__builtin_fabsf(a - med) < EPS) ? (a + wv.w) : 0.0f; m = fmaxf(m, v);
    }
    out[(size_t)b * n_out + o] = m;   // coalesced b32 store
  }
}

extern "C" void kernel_launch(void* const* d_in, const int* in_sizes, int n_in,
                              void* d_out, int out_size, void* d_ws, size_t ws_size,
                              hipStream_t stream) {
  const float* x       = (const float*)d_in[0];
  const float* weights = (const float*)d_in[1];
  const int*   syn_idx = (const int*)d_in[2];
  float* out = (float*)d_out;

  const int batch = in_sizes[0] / N_IN;   // 2048
  const int n_out = in_sizes[1] / DEG;    // 4096

  hipLaunchKernelGGL(boreal_kernel, dim3(batch), dim3(TPB), 0, stream,
                     x, weights, syn_idx, out, n_out);
}